// non_PyG_GAT_77618648973851
// MI455X (gfx1250) — compile-verified
//
#include <hip/hip_runtime.h>
#include <math.h>

#define NNODES 100000
#define NEDGES 3200000
#define FIN    500
#define H64    64      // heads*hid layer1
#define NC     41
#define NCP    48      // padded stride for layer-2 features

typedef __attribute__((ext_vector_type(2))) float v2f;
typedef __attribute__((ext_vector_type(8))) float v8f;

// ---------------- utility ----------------

__global__ void fill_kernel(float* __restrict__ p, float v, long n) {
  long i = (long)blockIdx.x * blockDim.x + threadIdx.x;
  if (i < n) p[i] = v;
}

__device__ __forceinline__ void atomicMaxF(float* addr, float v) {
  // monotonic bit-pattern trick; works with -inf init
  if (v >= 0.0f) atomicMax((int*)addr, __float_as_int(v));
  else           atomicMin((unsigned int*)addr, __float_as_uint(v));
}

// ---------------- GEMM 1: X[N,500] @ W1[500,64] -> H1[N,64] (fp32 WMMA) --------
// LDS holds W1 in K-pair interleaved layout:
//   sWp[pair*128 + 2*n + j] = W1[2*pair + j][n]   (pair = k/2, j in {0,1})
// so a WMMA B fragment {W[k][n], W[k+1][n]} is ONE contiguous ds_load_b64.

__global__ __launch_bounds__(256) void gemm1_kernel(const float* __restrict__ X,
                                                    const float* __restrict__ W,
                                                    float* __restrict__ H) {
  __shared__ float sWp[(FIN / 2) * 128];            // 125 KB
  // coalesced global reads; strided LDS writes (one-time fill)
  for (int i = threadIdx.x; i < FIN * H64; i += 256) {
    int pair = i >> 7;            // i = pair*128 + j*64 + n
    int j    = (i >> 6) & 1;
    int n    = i & 63;
    sWp[pair * 128 + 2 * n + j] = W[(2 * pair + j) * H64 + n];
  }
  __syncthreads();

  const int wave = threadIdx.x >> 5;
  const int lane = threadIdx.x & 31;
  const long row0 = (long)blockIdx.x * 128 + wave * 16;
  if (row0 >= NNODES) return;                       // whole-wave uniform exit

  const int lo = lane & 15;                         // M (A) / N (B,C) position
  const int hi = lane >> 4;                         // K-half selector
  const float* __restrict__ xp = X + (row0 + lo) * FIN + hi * 2;

  v8f acc0 = {}, acc1 = {}, acc2 = {}, acc3 = {};
  for (int k0 = 0; k0 < FIN; k0 += 4) {
    v2f a = *(const v2f*)(xp + k0);                 // K = k0+2*hi, k0+2*hi+1
    const float* bp = sWp + (k0 / 2 + hi) * 128 + 2 * lo;
    v2f b0 = *(const v2f*)(bp + 0);
    v2f b1 = *(const v2f*)(bp + 32);
    v2f b2 = *(const v2f*)(bp + 64);
    v2f b3 = *(const v2f*)(bp + 96);
    acc0 = __builtin_amdgcn_wmma_f32_16x16x4_f32(false, a, false, b0, (short)0, acc0, false, false);
    acc1 = __builtin_amdgcn_wmma_f32_16x16x4_f32(false, a, false, b1, (short)0, acc1, false, false);
    acc2 = __builtin_amdgcn_wmma_f32_16x16x4_f32(false, a, false, b2, (short)0, acc2, false, false);
    acc3 = __builtin_amdgcn_wmma_f32_16x16x4_f32(false, a, false, b3, (short)0, acc3, false, false);
  }
  float* __restrict__ hp = H + (row0 + hi * 8) * H64 + lo;
#pragma unroll
  for (int v = 0; v < 8; ++v) {
    hp[(long)v * H64 + 0]  = acc0[v];
    hp[(long)v * H64 + 16] = acc1[v];
    hp[(long)v * H64 + 32] = acc2[v];
    hp[(long)v * H64 + 48] = acc3[v];
  }
}

// ---------------- GEMM 2: Hin[N,64] @ W2[64,41] -> H2[N,48-pad] (fp32 WMMA) ----
// Same K-pair interleaved LDS layout, stride 2*NCP = 96, zero-padded cols.

__global__ __launch_bounds__(256) void gemm2_kernel(const float* __restrict__ Hin,
                                                    const float* __restrict__ W2,
                                                    float* __restrict__ H2) {
  __shared__ float sWp[(H64 / 2) * 2 * NCP];        // 12 KB
  for (int i = threadIdx.x; i < H64 * NCP; i += 256) {
    int pair = i / (2 * NCP);     // i = pair*96 + j*48 + c
    int r    = i % (2 * NCP);
    int j    = r / NCP;
    int c    = r % NCP;
    sWp[pair * (2 * NCP) + 2 * c + j] =
        (c < NC) ? W2[(2 * pair + j) * NC + c] : 0.0f;
  }
  __syncthreads();

  const int wave = threadIdx.x >> 5;
  const int lane = threadIdx.x & 31;
  const long row0 = (long)blockIdx.x * 128 + wave * 16;
  if (row0 >= NNODES) return;

  const int lo = lane & 15;
  const int hi = lane >> 4;
  const float* __restrict__ xp = Hin + (row0 + lo) * H64 + hi * 2;

  v8f acc0 = {}, acc1 = {}, acc2 = {};
#pragma unroll
  for (int k0 = 0; k0 < H64; k0 += 4) {
    v2f a = *(const v2f*)(xp + k0);
    const float* bp = sWp + (k0 / 2 + hi) * (2 * NCP) + 2 * lo;
    v2f b0 = *(const v2f*)(bp + 0);
    v2f b1 = *(const v2f*)(bp + 32);
    v2f b2 = *(const v2f*)(bp + 64);
    acc0 = __builtin_amdgcn_wmma_f32_16x16x4_f32(false, a, false, b0, (short)0, acc0, false, false);
    acc1 = __builtin_amdgcn_wmma_f32_16x16x4_f32(false, a, false, b1, (short)0, acc1, false, false);
    acc2 = __builtin_amdgcn_wmma_f32_16x16x4_f32(false, a, false, b2, (short)0, acc2, false, false);
  }
  // pad cols 41..47 multiply zero weights -> store unconditionally
  float* __restrict__ hp = H2 + (row0 + hi * 8) * NCP + lo;
#pragma unroll
  for (int v = 0; v < 8; ++v) {
    hp[(long)v * NCP + 0]  = acc0[v];
    hp[(long)v * NCP + 16] = acc1[v];
    hp[(long)v * NCP + 32] = acc2[v];
  }
}

// ---------------- per-node attention halves -----------------------------------

__global__ void attn1_kernel(const float* __restrict__ H1,
                             const float* __restrict__ a1s, const float* __restrict__ a1d,
                             float* __restrict__ asrc, float* __restrict__ adst) {
  long t = (long)blockIdx.x * blockDim.x + threadIdx.x;
  if (t >= (long)NNODES * 8) return;
  long n = t >> 3; int h = (int)(t & 7);
  const float* hv = H1 + n * H64 + h * 8;
  const float* vs = a1s + h * 8;
  const float* vd = a1d + h * 8;
  float s1 = 0.f, s2 = 0.f;
#pragma unroll
  for (int d = 0; d < 8; ++d) { float x = hv[d]; s1 += x * vs[d]; s2 += x * vd[d]; }
  asrc[t] = s1; adst[t] = s2;
}

__global__ void attn2_kernel(const float* __restrict__ H2,
                             const float* __restrict__ a2s, const float* __restrict__ a2d,
                             float* __restrict__ asrc, float* __restrict__ adst) {
  long n = (long)blockIdx.x * blockDim.x + threadIdx.x;
  if (n >= NNODES) return;
  const float* hv = H2 + n * NCP;
  float s1 = 0.f, s2 = 0.f;
  for (int c = 0; c < NC; ++c) { float x = hv[c]; s1 += x * a2s[c]; s2 += x * a2d[c]; }
  asrc[n] = s1; adst[n] = s2;
}

// ---------------- edge passes, layer 1 (8 heads) ------------------------------

__global__ void edge_max1_kernel(const long long* __restrict__ ei,
                                 const float* __restrict__ asrc, const float* __restrict__ adst,
                                 float* __restrict__ emax) {
  long e = (long)blockIdx.x * blockDim.x + threadIdx.x;
  if (e >= NEDGES) return;
  long s = (long)ei[e], d = (long)ei[(long)NEDGES + e];
  const float* as = asrc + s * 8;
  const float* ad = adst + d * 8;
  float* em = emax + d * 8;
#pragma unroll
  for (int h = 0; h < 8; ++h) {
    float v = as[h] + ad[h];
    v = v > 0.f ? v : 0.2f * v;                     // leaky_relu(0.2)
    atomicMaxF(em + h, v);
  }
}

__global__ void edge_sum1_kernel(const long long* __restrict__ ei,
                                 const float* __restrict__ asrc, const float* __restrict__ adst,
                                 const float* __restrict__ emax, const float* __restrict__ H1,
                                 float* __restrict__ denom, float* __restrict__ agg) {
  long e = (long)blockIdx.x * blockDim.x + threadIdx.x;
  if (e >= NEDGES) return;
  long s = (long)ei[e], d = (long)ei[(long)NEDGES + e];
  const float* as = asrc + s * 8;
  const float* ad = adst + d * 8;
  const float* em = emax + d * 8;
  const float* hs = H1 + s * H64;
  float* ag = agg + d * H64;
#pragma unroll
  for (int h = 0; h < 8; ++h) {
    float v = as[h] + ad[h];
    v = v > 0.f ? v : 0.2f * v;
    float w = expf(v - em[h]);
    atomicAdd(denom + d * 8 + h, w);
#pragma unroll
    for (int t = 0; t < 8; ++t)
      atomicAdd(ag + h * 8 + t, w * hs[h * 8 + t]);
  }
}

__global__ void finalize1_kernel(const float* __restrict__ agg, const float* __restrict__ denom,
                                 float* __restrict__ Hout) {
  long t = (long)blockIdx.x * blockDim.x + threadIdx.x;
  if (t >= (long)NNODES * H64) return;
  long n = t >> 6; int c = (int)(t & 63); int h = c >> 3;
  float v = agg[t] / (denom[n * 8 + h] + 1e-16f);
  Hout[t] = v > 0.f ? v : (expf(v) - 1.0f);         // ELU(alpha=1)
}

// ---------------- edge passes, layer 2 (1 head, 41 ch, stride 48) -------------

__global__ void edge_max2_kernel(const long long* __restrict__ ei,
                                 const float* __restrict__ asrc, const float* __restrict__ adst,
                                 float* __restrict__ emax) {
  long e = (long)blockIdx.x * blockDim.x + threadIdx.x;
  if (e >= NEDGES) return;
  long s = (long)ei[e], d = (long)ei[(long)NEDGES + e];
  float v = asrc[s] + adst[d];
  v = v > 0.f ? v : 0.2f * v;
  atomicMaxF(emax + d, v);
}

__global__ void edge_sum2_kernel(const long long* __restrict__ ei,
                                 const float* __restrict__ asrc, const float* __restrict__ adst,
                                 const float* __restrict__ emax, const float* __restrict__ H2,
                                 float* __restrict__ denom, float* __restrict__ agg) {
  long e = (long)blockIdx.x * blockDim.x + threadIdx.x;
  if (e >= NEDGES) return;
  long s = (long)ei[e], d = (long)ei[(long)NEDGES + e];
  float v = asrc[s] + adst[d];
  v = v > 0.f ? v : 0.2f * v;
  float w = expf(v - emax[d]);
  atomicAdd(denom + d, w);
  const float* hs = H2 + s * NCP;
  float* ag = agg + d * NCP;
  for (int c = 0; c < NC; ++c)
    atomicAdd(ag + c, w * hs[c]);
}

__global__ void final_kernel(const float* __restrict__ agg, const float* __restrict__ denom,
                             float* __restrict__ out) {
  long n = (long)blockIdx.x * blockDim.x + threadIdx.x;
  if (n >= NNODES) return;
  const float* a = agg + n * NCP;
  float dn = denom[n] + 1e-16f;
  float m = -__builtin_huge_valf();
  for (int c = 0; c < NC; ++c) m = fmaxf(m, a[c] / dn);
  float sum = 0.f;
  for (int c = 0; c < NC; ++c) sum += expf(a[c] / dn - m);
  float lse = m + logf(sum);
  float* o = out + n * NC;
  for (int c = 0; c < NC; ++c) o[c] = a[c] / dn - lse;
}

// ---------------- host launch -------------------------------------------------

extern "C" void kernel_launch(void* const* d_in, const int* in_sizes, int n_in,
                              void* d_out, int out_size, void* d_ws, size_t ws_size,
                              hipStream_t stream) {
  const float*     x   = (const float*)d_in[0];
  const long long* ei  = (const long long*)d_in[1];   // int64 edge_index [2,E]
  const float*     W1  = (const float*)d_in[2];
  const float*     a1s = (const float*)d_in[3];
  const float*     a1d = (const float*)d_in[4];
  const float*     W2  = (const float*)d_in[5];
  const float*     a2s = (const float*)d_in[6];
  const float*     a2d = (const float*)d_in[7];
  float*           out = (float*)d_out;
  float*           ws  = (float*)d_ws;

  // workspace layout (floats); buffers are reused across layers
  float* H1  = ws;                  // N*64       ; later agg2 (N*48)
  float* as1 = ws + 6400000L;       // N*8        ; later asrc2 (N)
  float* ad1 = ws + 7200000L;       // N*8        ; later adst2 (N)
  float* em1 = ws + 8000000L;       // N*8        ; later emax2 (N)
  float* dn1 = ws + 8800000L;       // N*8        ; later denom2 (N)
  float* ag1 = ws + 9600000L;       // N*64       ; later H2 (N*48)

  const float NEG_INF = -__builtin_huge_valf();
  const int TB = 256;
  const int gN8  = (NNODES * 8 + TB - 1) / TB;
  const int gN64 = (NNODES * H64 + TB - 1) / TB;
  const int gN   = (NNODES + TB - 1) / TB;
  const int gE   = (NEDGES + TB - 1) / TB;
  const int gRow = (NNODES + 127) / 128;            // 782 row-blocks of 128

  // ---- layer 1 ----
  fill_kernel<<<gN8, TB, 0, stream>>>(em1, NEG_INF, (long)NNODES * 8);
  fill_kernel<<<gN8, TB, 0, stream>>>(dn1, 0.0f, (long)NNODES * 8);
  fill_kernel<<<gN64, TB, 0, stream>>>(ag1, 0.0f, (long)NNODES * H64);

  gemm1_kernel<<<gRow, TB, 0, stream>>>(x, W1, H1);
  attn1_kernel<<<gN8, TB, 0, stream>>>(H1, a1s, a1d, as1, ad1);
  edge_max1_kernel<<<gE, TB, 0, stream>>>(ei, as1, ad1, em1);
  edge_sum1_kernel<<<gE, TB, 0, stream>>>(ei, as1, ad1, em1, H1, dn1, ag1);
  finalize1_kernel<<<gN64, TB, 0, stream>>>(ag1, dn1, H1);   // H1 := elu(out1)

  // ---- layer 2 ----
  float* H2   = ag1;                // overwritten by gemm2 (agg1 no longer needed)
  float* agg2 = H1;                 // H1e free after gemm2 reads it
  gemm2_kernel<<<gRow, TB, 0, stream>>>(H1, W2, H2);

  const int gN48 = (NNODES * NCP + TB - 1) / TB;
  fill_kernel<<<gN48, TB, 0, stream>>>(agg2, 0.0f, (long)NNODES * NCP);
  fill_kernel<<<gN, TB, 0, stream>>>(em1, NEG_INF, (long)NNODES);
  fill_kernel<<<gN, TB, 0, stream>>>(dn1, 0.0f, (long)NNODES);

  attn2_kernel<<<gN, TB, 0, stream>>>(H2, a2s, a2d, as1, ad1);
  edge_max2_kernel<<<gE, TB, 0, stream>>>(ei, as1, ad1, em1);
  edge_sum2_kernel<<<gE, TB, 0, stream>>>(ei, as1, ad1, em1, H2, dn1, agg2);
  final_kernel<<<gN, TB, 0, stream>>>(agg2, dn1, out);
}